// DDE_6081673691476
// MI455X (gfx1250) — compile-verified
//
#include <hip/hip_runtime.h>
#include <stdint.h>

// ---------------------------------------------------------------------------
// Mean-aggregation message passing (PyG MessagePassing aggr='mean'), 4 rounds:
//   2 rounds over edge_index, 2 rounds over reverse_edge_index.
// Roofline: node table is 6.4 MB (fits in 192 MB L2); dominant cost is the
// edge-index stream (25.6 MB/round from HBM) + ~51M L2-resident f32 atomics
// per round. gfx1250-specific paths: GLOBAL_LOAD_ASYNC_TO_LDS_B32 double-
// buffered staging of edge indices, S_WAIT_ASYNCCNT split waits, native
// GLOBAL_ATOMIC_ADD_F32 scatters, GLOBAL_PREFETCH_B8 on the index stream.
// ---------------------------------------------------------------------------

#define NDIM    16
#define CHUNK   256
#define THREADS 256

// ---- s_wait_asynccnt ------------------------------------------------------
#if defined(__HIP_DEVICE_COMPILE__) && __has_builtin(__builtin_amdgcn_s_wait_asynccnt)
#define WAIT_ASYNC(n) __builtin_amdgcn_s_wait_asynccnt(n)
#else
#define WAIT_ASYNC(n) asm volatile("s_wait_asynccnt %0" ::"i"(n) : "memory")
#endif

// ---- async global -> LDS copy of one dword per lane -----------------------
__device__ __forceinline__ void async_ld_b32(const int* g, int* l) {
#if defined(__HIP_DEVICE_COMPILE__) && __has_builtin(__builtin_amdgcn_global_load_async_to_lds_b32)
  __builtin_amdgcn_global_load_async_to_lds_b32(
      (__attribute__((address_space(1))) int*)(int*)g,
      (__attribute__((address_space(3))) int*)l,
      /*offset=*/0, /*cpol=*/0);
#else
  *l = *g;   // synchronous fallback (still correct)
#endif
}

// ---- native f32 atomic add ------------------------------------------------
__device__ __forceinline__ void atom_add_f32(float* p, float v) {
  unsafeAtomicAdd(p, v);  // global_atomic_add_f32 on gfx1250
}

// ---------------------------------------------------------------------------
// Scatter-add: for each edge e, sum[dst[e]][:] += x[src[e]][:].
// Edge indices are staged into LDS with double-buffered async copies so the
// index stream for chunk k+1 is in flight while chunk k's atomics drain.
// Each thread only touches its own LDS slot, so per-wave ASYNCcnt ordering
// (loads complete in order) is sufficient -- no workgroup barrier needed.
// Defined FIRST so the disasm snippet shows the hot loop.
// ---------------------------------------------------------------------------
__global__ void __launch_bounds__(THREADS)
scatter_sum_kernel(const float* __restrict__ x,
                   const int* __restrict__ src,
                   const int* __restrict__ dst,
                   float* __restrict__ sum,
                   int E, int cpb, int nchunks)
{
  __shared__ int s_src[2][CHUNK];
  __shared__ int s_dst[2][CHUNK];

  const int tid = threadIdx.x;
  const int c0  = blockIdx.x * cpb;
  if (c0 >= nchunks) return;
  const int c1 = (c0 + cpb < nchunks) ? (c0 + cpb) : nchunks;

  // Prime buffer 0 with the first chunk (clamped addresses keep EXEC all-1s).
  {
    int e  = c0 * CHUNK + tid;
    int ec = (e < E) ? e : (E - 1);
    async_ld_b32(src + ec, &s_src[0][tid]);
    async_ld_b32(dst + ec, &s_dst[0][tid]);
  }

  int buf = 0;
  for (int c = c0; c < c1; ++c) {
    if (c + 1 < c1) {
      // Kick off next chunk before consuming the current one.
      int e  = (c + 1) * CHUNK + tid;
      int ec = (e < E) ? e : (E - 1);
      async_ld_b32(src + ec, &s_src[buf ^ 1][tid]);
      async_ld_b32(dst + ec, &s_dst[buf ^ 1][tid]);
      // Async loads complete in order: <=2 outstanding => current chunk done.
      WAIT_ASYNC(2);
    } else {
      WAIT_ASYNC(0);
    }

    int e = c * CHUNK + tid;
    if (e < E) {
      int sidx = s_src[buf][tid];   // ds_load_b32
      int didx = s_dst[buf][tid];   // ds_load_b32
      const float4* xs = (const float4*)(x + (size_t)sidx * NDIM);
      float4 a = xs[0], b = xs[1], cc = xs[2], d = xs[3];  // global_load_b128 x4
      float* o = sum + (size_t)didx * NDIM;
      atom_add_f32(o + 0,  a.x);  atom_add_f32(o + 1,  a.y);
      atom_add_f32(o + 2,  a.z);  atom_add_f32(o + 3,  a.w);
      atom_add_f32(o + 4,  b.x);  atom_add_f32(o + 5,  b.y);
      atom_add_f32(o + 6,  b.z);  atom_add_f32(o + 7,  b.w);
      atom_add_f32(o + 8,  cc.x); atom_add_f32(o + 9,  cc.y);
      atom_add_f32(o + 10, cc.z); atom_add_f32(o + 11, cc.w);
      atom_add_f32(o + 12, d.x);  atom_add_f32(o + 13, d.y);
      atom_add_f32(o + 14, d.z);  atom_add_f32(o + 15, d.w);
    }
    buf ^= 1;
  }
}

// ---------------------------------------------------------------------------
__global__ void __launch_bounds__(THREADS)
count_kernel(const int* __restrict__ dst, uint32_t* __restrict__ cnt, int E) {
  int i = blockIdx.x * blockDim.x + threadIdx.x;
  int stride = gridDim.x * blockDim.x;
  for (; i < E; i += stride) {
    __builtin_prefetch(dst + i + stride, 0, 0);  // global_prefetch_b8
    atomicAdd(&cnt[dst[i]], 1u);                 // global_atomic_add_u32
  }
}

// in-place: u32 count -> f32 reciprocal of max(count, 1)
__global__ void __launch_bounds__(THREADS)
inv_kernel(uint32_t* __restrict__ cnt_inv, int N) {
  int i = blockIdx.x * blockDim.x + threadIdx.x;
  if (i < N) {
    uint32_t c = cnt_inv[i];
    float f = 1.0f / (float)(c > 1u ? c : 1u);
    ((float*)cnt_inv)[i] = f;
  }
}

// grid-stride zero, 16B per thread per step (n must be a multiple of 4 dwords
// for the uint4 path; both users (N*16 and N rounded up by caller) satisfy it)
__global__ void __launch_bounds__(THREADS)
zero_kernel4(uint4* __restrict__ p, int nquads) {
  int i = blockIdx.x * blockDim.x + threadIdx.x;
  int stride = gridDim.x * blockDim.x;
  uint4 z = {0u, 0u, 0u, 0u};
  for (; i < nquads; i += stride) p[i] = z;
}

// out[n][:] = sum[n][:] * inv[n]; one thread per float4 (4 divides NDIM).
__global__ void __launch_bounds__(THREADS)
normalize_kernel(const float* __restrict__ sum, const float* __restrict__ inv,
                 float* __restrict__ out, int nquads) {
  int i = blockIdx.x * blockDim.x + threadIdx.x;
  if (i < nquads) {
    float4 v = ((const float4*)sum)[i];
    float s = inv[i >> 2];  // node = (i*4)/16
    v.x *= s; v.y *= s; v.z *= s; v.w *= s;
    ((float4*)out)[i] = v;
  }
}

// ---------------------------------------------------------------------------
extern "C" void kernel_launch(void* const* d_in, const int* in_sizes, int n_in,
                              void* d_out, int out_size, void* d_ws, size_t ws_size,
                              hipStream_t stream) {
  const float* x0  = (const float*)d_in[0];           // [N, 16] f32
  const int*   ei  = (const int*)d_in[1];             // [2, E] row-major
  const int*   rei = (const int*)d_in[2];             // [2, E]
  const int N = in_sizes[0] / NDIM;                   // 100000
  const int E = in_sizes[1] / 2;                      // 3200000
  float* outp = (float*)d_out;                        // [4, N, 16] f32

  float*    sum = (float*)d_ws;                                        // N*16 f32
  uint32_t* cnt = (uint32_t*)((char*)d_ws +
                              (size_t)N * NDIM * sizeof(float));       // N u32 -> f32 inv

  const int nelem    = N * NDIM;          // multiple of 16 dwords
  const int cntquads = (N + 3) / 4;       // zero N (rounded up) dwords of cnt
  const int nchunks  = (E + CHUNK - 1) / CHUNK;
  int nblocks = 1250;
  if (nblocks > nchunks) nblocks = nchunks;
  const int cpb = (nchunks + nblocks - 1) / nblocks;

  const int* edge_sets[2] = {ei, rei};

  for (int dir = 0; dir < 2; ++dir) {
    const int* s = edge_sets[dir];
    const int* d = s + E;  // dst row

    zero_kernel4<<<256, THREADS, 0, stream>>>((uint4*)cnt, cntquads);
    count_kernel<<<2048, THREADS, 0, stream>>>(d, cnt, E);
    inv_kernel<<<(N + THREADS - 1) / THREADS, THREADS, 0, stream>>>(cnt, N);

    const float* xin = x0;
    for (int r = 0; r < 2; ++r) {
      zero_kernel4<<<1024, THREADS, 0, stream>>>((uint4*)sum, nelem / 4);
      scatter_sum_kernel<<<nblocks, THREADS, 0, stream>>>(xin, s, d, sum, E, cpb, nchunks);
      float* slot = outp + (size_t)(dir * 2 + r) * nelem;
      int nquads = nelem / 4;
      normalize_kernel<<<(nquads + THREADS - 1) / THREADS, THREADS, 0, stream>>>(
          sum, (const float*)cnt, slot, nquads);
      xin = slot;  // next round consumes this round's normalized output
    }
  }
}